// CLOPLayer_74586402062828
// MI455X (gfx1250) — compile-verified
//
#include <hip/hip_runtime.h>
#include <stdint.h>

// CLOP layer: perm generation (JAX threefry + serial swap scan re-expressed as
// parallel backward transposition walk) + HBM-roofline gather.
// x: (512, 9, 84, 84) fp32. n = 7056, dim = 84, steps m = 2n-1 = 14111.

#define N_ELEM   7056
#define DIM      84
#define M_STEPS  14111
#define NPLANES  4
#define PLANES_TOTAL 4608   // 512*9
#define CHUNKS_MAX 32

typedef uint32_t v4u __attribute__((ext_vector_type(4)));
typedef int      v8i __attribute__((ext_vector_type(8)));
typedef int      v4i __attribute__((ext_vector_type(4)));

__device__ __forceinline__ uint32_t rotl32(uint32_t x, int r) {
    return (x << r) | (x >> (32 - r));
}

// ---------------------------------------------------------------------------
// Kernel 1: generate swap pairs (a_t, b_t) for every step t, in parallel.
// Exact replication of jax.random.choice(key(42), 5, p=probs):
//   threefry2x32 with key (0,42); counters arange(14111) padded odd->even,
//   split into halves (x0[k]=k, x1[k]=k+7056, last pad 0);
//   u = bitcast((bits>>9)|0x3f800000) - 1;  r = cuml[-1]*(1-u);
//   ind = searchsorted_left(cuml, r).
// Step t swaps positions (i, i+off[ind]) iff ind!=0 && 0<idx<n, where
// i = |t - (n-1)|, off = {0,+1,-1,+dim,-dim}. No-op encoded as a==b.
// ---------------------------------------------------------------------------
__global__ void __launch_bounds__(256)
clop_swaps_kernel(int2* __restrict__ swaps) {
    int t = blockIdx.x * blockDim.x + threadIdx.x;
    if (t >= M_STEPS) return;

    int k = (t < N_ELEM) ? t : (t - N_ELEM);
    uint32_t x0 = (uint32_t)k;
    uint32_t x1 = (k == N_ELEM - 1) ? 0u : (uint32_t)(k + N_ELEM);

    const uint32_t ks0 = 0u, ks1 = 42u;
    const uint32_t ks2 = ks0 ^ ks1 ^ 0x1BD11BDAu;
    x0 += ks0; x1 += ks1;
#define TF_R(r) { x0 += x1; x1 = rotl32(x1, (r)); x1 ^= x0; }
    TF_R(13) TF_R(15) TF_R(26) TF_R(6)   x0 += ks1; x1 += ks2 + 1u;
    TF_R(17) TF_R(29) TF_R(16) TF_R(24)  x0 += ks2; x1 += ks0 + 2u;
    TF_R(13) TF_R(15) TF_R(26) TF_R(6)   x0 += ks0; x1 += ks1 + 3u;
    TF_R(17) TF_R(29) TF_R(16) TF_R(24)  x0 += ks1; x1 += ks2 + 4u;
    TF_R(13) TF_R(15) TF_R(26) TF_R(6)   x0 += ks2; x1 += ks0 + 5u;
#undef TF_R
    uint32_t bits = (t < N_ELEM) ? x0 : x1;

    float u = __uint_as_float((bits >> 9) | 0x3f800000u) - 1.0f;

    const float c0 = 0.55f;
    const float c1 = c0 + 0.1125f;
    const float c2 = c1 + 0.1125f;
    const float c3 = c2 + 0.1125f;
    const float c4 = c3 + 0.1125f;
    float r = c4 * (1.0f - u);
    int ind = (int)(c0 < r) + (int)(c1 < r) + (int)(c2 < r) + (int)(c3 < r);

    int off = (ind == 0) ? 0 : (ind == 1) ? 1 : (ind == 2) ? -1
            : (ind == 3) ? DIM : -DIM;
    int i = (t >= N_ELEM - 1) ? (t - (N_ELEM - 1)) : ((N_ELEM - 1) - t);
    int idx = i + off;
    bool doswap = (ind != 0) && (idx > 0) && (idx < N_ELEM);
    int b = doswap ? idx : i;   // idx already in (0, n) when doswap
    swaps[t] = make_int2(i, b);
}

// ---------------------------------------------------------------------------
// Kernel 2a: per-chunk backward transposition walk.
// perm_final[i] = tau_1(tau_2(...tau_m(i))). Chunk c builds
// M_c[i] = tau_{first(c)}(...tau_{last(c)}(i)) by walking its step range
// from last to first; branch-free transposition application.
// Grid: C * 28 blocks of 256 (28*256 >= 7056).
// ---------------------------------------------------------------------------
__global__ void __launch_bounds__(256)
clop_chunk_walk_kernel(const int2* __restrict__ swaps, int* __restrict__ M,
                       int nchunks, int chunk_len) {
    const int blocks_per_chunk = (N_ELEM + 255) / 256;   // 28
    int c = blockIdx.x / blocks_per_chunk;
    int i = (blockIdx.x % blocks_per_chunk) * blockDim.x + threadIdx.x;
    if (c >= nchunks || i >= N_ELEM) return;

    int t_lo = c * chunk_len;
    int t_hi = t_lo + chunk_len;
    if (t_hi > M_STEPS) t_hi = M_STEPS;

    int j = i;
    for (int t = t_hi - 1; t >= t_lo; --t) {
        int2 s = swaps[t];                 // wave-uniform address -> s_load
        j = (j == s.x) ? s.y : ((j == s.y) ? s.x : j);
    }
    M[c * N_ELEM + i] = j;
}

// Kernel 2b: compose the chunk maps: perm[i] = M_0[M_1[...M_{C-1}[i]]].
__global__ void __launch_bounds__(256)
clop_compose_kernel(const int* __restrict__ M, int* __restrict__ perm,
                    int nchunks) {
    int i = blockIdx.x * blockDim.x + threadIdx.x;
    if (i >= N_ELEM) return;
    int j = i;
    for (int c = nchunks - 1; c >= 0; --c)
        j = M[c * N_ELEM + j];             // tiny arrays, L2-hot
    perm[i] = j;
}

// ---------------------------------------------------------------------------
// Kernel 3: the roofline gather. Each block handles NPLANES planes.
// Wave 0 DMAs perm (7056 dwords = 28224 B) into LDS with the Tensor Data
// Mover, waits on TENSORcnt, then all waves gather with int4 LDS reads,
// near-coalesced b32 global reads, and coalesced float4 (b128) stores.
// ---------------------------------------------------------------------------
__global__ void __launch_bounds__(256)
clop_gather_kernel(const float* __restrict__ x, const int* __restrict__ perm,
                   float* __restrict__ out) {
    __shared__ int lperm[N_ELEM];

    if (threadIdx.x < 32) {   // one wave issues the TDM op (EXEC-independent)
#if defined(__has_builtin) && __has_builtin(__builtin_amdgcn_tensor_load_to_lds)
        uint32_t lds_addr = (uint32_t)(uintptr_t)(&lperm[0]);
        uint64_t ga = (uint64_t)(uintptr_t)perm;
        v4u g0;
        g0[0] = 1u;                               // count=1, user descriptor
        g0[1] = lds_addr;                         // lds_addr
        g0[2] = (uint32_t)ga;                     // global_addr[31:0]
        g0[3] = (uint32_t)(ga >> 32) | (2u << 30);// global_addr[56:32] | type=2
        v8i g1;
        g1[0] = (int)(2u << 16);                  // data_size = 4 bytes
        g1[1] = (int)((uint32_t)N_ELEM << 16);    // tensor_dim0 (low 16)
        g1[2] = 0;                                // tensor_dim0 hi / dim1 lo
        g1[3] = (int)((uint32_t)N_ELEM << 16);    // tile_dim0
        g1[4] = 0;                                // tile_dim1/2 unused (1D)
        g1[5] = N_ELEM;                           // tensor_dim0_stride lo
        g1[6] = 0;
        g1[7] = 0;
        v4i gz4 = {0, 0, 0, 0};                   // <=2D: groups 2/3 unused
        v8i gz8 = {0, 0, 0, 0, 0, 0, 0, 0};
        __builtin_amdgcn_tensor_load_to_lds(g0, g1, gz4, gz4, gz8, 0);
        __builtin_amdgcn_s_wait_tensorcnt(0);
#else
        for (int i = (int)threadIdx.x; i < N_ELEM; i += 32)
            lperm[i] = perm[i];
#endif
    }
    __syncthreads();

    const int plane0 = blockIdx.x * NPLANES;
    const int4* lp = (const int4*)lperm;
    for (int p = 0; p < NPLANES; ++p) {
        const long long base = (long long)(plane0 + p) * N_ELEM;
        const float* xp = x + base;
        float4* op = (float4*)(out + base);
        for (int i = (int)threadIdx.x; i < N_ELEM / 4; i += (int)blockDim.x) {
            int4 pm = lp[i];          // ds_load_b128
            float4 v;
            v.x = xp[pm.x];
            v.y = xp[pm.y];
            v.z = xp[pm.z];
            v.w = xp[pm.w];
            op[i] = v;                // global_store_b128, coalesced
        }
    }
}

// ---------------------------------------------------------------------------
extern "C" void kernel_launch(void* const* d_in, const int* in_sizes, int n_in,
                              void* d_out, int out_size, void* d_ws, size_t ws_size,
                              hipStream_t stream) {
    const float* x = (const float*)d_in[0];
    float* out = (float*)d_out;

    // Workspace layout: [swaps: 14111*int2][M: C*7056*int][perm: 7056*int]
    const size_t swaps_bytes = (size_t)M_STEPS * sizeof(int2);  // 112888
    const size_t swaps_off_end = (swaps_bytes + 15) & ~(size_t)15;

    int nchunks = CHUNKS_MAX;
    size_t need = swaps_off_end + (size_t)(nchunks + 1) * N_ELEM * sizeof(int);
    if (ws_size < need) {
        nchunks = 1;  // minimal-footprint fallback (slower perm build)
    }
    int chunk_len = (M_STEPS + nchunks - 1) / nchunks;

    int2* swaps = (int2*)d_ws;
    int*  M     = (int*)((char*)d_ws + swaps_off_end);
    int*  perm  = M + (size_t)nchunks * N_ELEM;

    // 1) parallel RNG + swap-pair generation
    clop_swaps_kernel<<<(M_STEPS + 255) / 256, 256, 0, stream>>>(swaps);

    // 2) chunked backward walks, then composition -> perm
    const int blocks_per_chunk = (N_ELEM + 255) / 256;   // 28
    clop_chunk_walk_kernel<<<nchunks * blocks_per_chunk, 256, 0, stream>>>(
        swaps, M, nchunks, chunk_len);
    clop_compose_kernel<<<blocks_per_chunk, 256, 0, stream>>>(M, perm, nchunks);

    // 3) HBM-roofline gather (TDM-staged perm in LDS)
    clop_gather_kernel<<<PLANES_TOTAL / NPLANES, 256, 0, stream>>>(x, perm, out);
}